// SpatioTemporalSampler_63737314673105
// MI455X (gfx1250) — compile-verified
//
#include <hip/hip_runtime.h>
#include <stdint.h>

// ---------------------------------------------------------------------------
// SpatioTemporalSampler for MI455X (gfx1250, wave32).
//
// Cost model: compute-bound on ~67M threefry2x32-20 hashes (neg mask ~98%
// dense); node/time ids (8MB) are L2-resident so the 64x reuse across batch
// rows is free. Data staging uses the CDNA5 Tensor Data Mover
// (tensor_load_to_lds + s_wait_tensorcnt) to DMA each 16KB id tile into LDS.
// ---------------------------------------------------------------------------

#define N_TOTAL  1048576
#define N_NODE   2048
#define BATCH    64
#define CHUNK    4096          // ids per block tile (2 x 16KB LDS)
#define THREADS  256           // 8 wave32 waves
#define HALF_CNT 33554432u     // threefry count-array half size for (64, 2^20)

// ---------------- threefry2x32 (20 rounds), bit-exact with JAX ----------------
#define TFROUND(x0, x1, r)                                                     \
  { x0 += x1; x1 = ((x1 << (r)) | (x1 >> (32 - (r)))); x1 ^= x0; }

__host__ __device__ __forceinline__ void threefry2x32(uint32_t ks0, uint32_t ks1,
                                                      uint32_t x0, uint32_t x1,
                                                      uint32_t& o0, uint32_t& o1) {
  uint32_t ks2 = ks0 ^ ks1 ^ 0x1BD11BDAu;
  x0 += ks0; x1 += ks1;
  TFROUND(x0, x1, 13) TFROUND(x0, x1, 15) TFROUND(x0, x1, 26) TFROUND(x0, x1, 6)
  x0 += ks1; x1 += ks2 + 1u;
  TFROUND(x0, x1, 17) TFROUND(x0, x1, 29) TFROUND(x0, x1, 16) TFROUND(x0, x1, 24)
  x0 += ks2; x1 += ks0 + 2u;
  TFROUND(x0, x1, 13) TFROUND(x0, x1, 15) TFROUND(x0, x1, 26) TFROUND(x0, x1, 6)
  x0 += ks0; x1 += ks1 + 3u;
  TFROUND(x0, x1, 17) TFROUND(x0, x1, 29) TFROUND(x0, x1, 16) TFROUND(x0, x1, 24)
  x0 += ks1; x1 += ks2 + 4u;
  TFROUND(x0, x1, 13) TFROUND(x0, x1, 15) TFROUND(x0, x1, 26) TFROUND(x0, x1, 6)
  x0 += ks2; x1 += ks0 + 5u;
  o0 = x0; o1 = x1;
}

// Candidate ranking key for argmax-of-uniform with first-occurrence ties:
// uniform is monotone in (bits >> 9); equal mantissa => smaller j wins.
__device__ __forceinline__ unsigned long long cand_key(uint32_t k0, uint32_t k1,
                                                       uint32_t i, uint32_t j) {
  uint32_t m = (i << 20) | j;          // flat index into (64, 2^20) uniform array
  uint32_t x0, x1, r0, r1;
  if (m < HALF_CNT) { x0 = m; x1 = m + HALF_CNT; }
  else              { x0 = m - HALF_CNT; x1 = m; }
  threefry2x32(k0, k1, x0, x1, r0, r1);
  uint32_t bits = (m < HALF_CNT) ? r0 : r1;
  uint32_t mant = bits >> 9;           // 23-bit mantissa, monotone with u
  return ((unsigned long long)mant << 32) | (unsigned long long)(0xFFFFFFFFu - j);
}

// ---------------- CDNA5 Tensor Data Mover: 1-D DWORD tile -> LDS ----------------
typedef unsigned int u32x4 __attribute__((ext_vector_type(4)));
typedef int          i32x4 __attribute__((ext_vector_type(4)));
typedef int          i32x8 __attribute__((ext_vector_type(8)));

__device__ __forceinline__ void tdm_load_1d_dw(uint32_t lds_off, const void* gptr,
                                               uint32_t ndw) {
  uint64_t ga   = (uint64_t)(uintptr_t)gptr;
  uint32_t galo = (uint32_t)__builtin_amdgcn_readfirstlane((int)(uint32_t)ga);
  uint32_t gahi = (uint32_t)__builtin_amdgcn_readfirstlane((int)(uint32_t)(ga >> 32));
  uint32_t ldo  = (uint32_t)__builtin_amdgcn_readfirstlane((int)lds_off);
  // D# group0: count=1 | lds_addr | global_addr[56:0] | type=2 (bits 127:126)
  u32x4 g0 = { 1u, ldo, galo, (gahi & 0x01FFFFFFu) | 0x80000000u };
  // D# group1: data_size=4B; tensor_dim0=ndw (bits 79:48); tensor_dim1=1
  // (bits 111:80); tile_dim0=ndw (bits 127:112); tile_dim1/2=0 (unused);
  // tensor_dim0_stride=ndw (bits 207:160).
  i32x8 g1 = { (int)(2u << 16),
               (int)((ndw & 0xFFFFu) << 16),
               (int)((ndw >> 16) | (1u << 16)),
               (int)(ndw << 16),
               0,
               (int)ndw,
               0, 0 };
  i32x4 gz4 = { 0, 0, 0, 0 };
  i32x8 gz8 = { 0, 0, 0, 0, 0, 0, 0, 0 };
  // 6-arg toolchain variant: (group0, group1, group2, group3, extra, cpol)
  __builtin_amdgcn_tensor_load_to_lds(g0, g1, gz4, gz4, gz8, 0);
}

// ---------------- Phase A: batch metadata + adjacency bitmask ----------------
__global__ __launch_bounds__(64) void sampler_prep(
    const int* __restrict__ node_ids, const int* __restrict__ time_ids,
    const float* __restrict__ adj, const int* __restrict__ batch_idx,
    int* __restrict__ meta, uint32_t* __restrict__ nbits,
    unsigned long long* __restrict__ cells) {
  const int i = blockIdx.x;     // batch row 0..63
  const int t = threadIdx.x;    // 0..63
  __shared__ int s_node_i;
  if (t == 0) {
    int bi = batch_idx[i];
    int ni = node_ids[bi];
    meta[i * 3 + 0] = ni;
    meta[i * 3 + 1] = time_ids[bi];
    meta[i * 3 + 2] = bi;
    s_node_i = ni;
  }
  if (t < 3) cells[i * 3 + t] = 0ull;   // zero argmax accumulators every call
  __syncthreads();
  const float* row = adj + (size_t)s_node_i * N_NODE + t * 32;
  uint32_t w = 0u;
  #pragma unroll
  for (int b = 0; b < 32; ++b) w |= (row[b] > 0.0f ? 1u : 0u) << b;
  nbits[i * 64 + t] = w;                // adj row as 2048-bit mask (exact: adj is 0/1)
}

// ---------------- Phase B: main scan (one block = one (row, j-tile)) ----------------
__global__ __launch_bounds__(THREADS) void sampler_main(
    const int* __restrict__ node_ids, const int* __restrict__ time_ids,
    const int* __restrict__ meta, const uint32_t* __restrict__ nbits,
    unsigned long long* __restrict__ cells,
    uint32_t kp1a, uint32_t kp1b, uint32_t kp2a, uint32_t kp2b,
    uint32_t kna, uint32_t knb) {
  __shared__ int s_node[CHUNK];
  __shared__ int s_time[CHUNK];
  __shared__ uint32_t s_bits[N_NODE / 32];
  __shared__ unsigned long long s_best[3];

  const int i    = blockIdx.y;
  const int base = blockIdx.x * CHUNK;
  const int tid  = threadIdx.x;

  const int node_i = meta[i * 3 + 0];
  const int time_i = meta[i * 3 + 1];
  const int bi     = meta[i * 3 + 2];

  if (tid < N_NODE / 32) s_bits[tid] = nbits[i * (N_NODE / 32) + tid];
  if (tid < 3) s_best[tid] = 0ull;

  if (tid == 0) {
    // TDM DMA of both id tiles into LDS (EXEC ignored -> single issue each),
    // then drain TENSORcnt before releasing the workgroup barrier.
    tdm_load_1d_dw((uint32_t)(uintptr_t)&s_node[0], node_ids + base, CHUNK);
    tdm_load_1d_dw((uint32_t)(uintptr_t)&s_time[0], time_ids + base, CHUNK);
    __builtin_amdgcn_s_wait_tensorcnt(0);
  }
  asm volatile("" ::: "memory");   // LDS now holds DMA data; don't fold loads
  __syncthreads();

  volatile const int* vn = s_node;
  volatile const int* vt = s_time;

  unsigned long long b1 = 0ull, b2 = 0ull, bn = 0ull;
  for (int k = tid; k < CHUNK; k += THREADS) {
    const uint32_t j = (uint32_t)(base + k);
    const int nid = vn[k];
    const int tt  = vt[k];
    int dt = tt - time_i; dt = (dt < 0) ? -dt : dt;
    const bool same    = (nid == node_i);
    const bool notself = ((int)j != bi);
    const bool neigh   = ((s_bits[nid >> 5] >> (nid & 31)) & 1u) != 0u;

    if (same & (dt <= 2) & notself) {                     // mask_pos1 (rare)
      unsigned long long c = cand_key(kp1a, kp1b, (uint32_t)i, j);
      if (c > b1) b1 = c;
    }
    if (neigh & (dt == 0)) {                              // mask_pos2 (rare)
      unsigned long long c = cand_key(kp2a, kp2b, (uint32_t)i, j);
      if (c > b2) b2 = c;
    }
    if ((same & (dt >= 12) & notself) | ((!neigh) & (dt > 2))) {  // mask_neg (~dense)
      unsigned long long c = cand_key(kna, knb, (uint32_t)i, j);
      if (c > bn) bn = c;
    }
  }

  if (b1) atomicMax(&s_best[0], b1);     // ds_max_u64 block reduction
  if (b2) atomicMax(&s_best[1], b2);
  if (bn) atomicMax(&s_best[2], bn);
  __syncthreads();
  if (tid < 3) {
    unsigned long long v = s_best[tid];
    if (v) atomicMax(&cells[i * 3 + tid], v);   // global_atomic_max_u64, 3/block
  }
}

// ---------------- Phase C: decode winners ----------------
__global__ void sampler_fin(const unsigned long long* __restrict__ cells,
                            float* __restrict__ out) {
  int t = blockIdx.x * blockDim.x + threadIdx.x;
  if (t < BATCH * 3) {
    unsigned long long v = cells[t];
    out[t] = v ? (float)(0xFFFFFFFFu - (uint32_t)v) : -1.0f;   // empty mask -> -1
  }
}

// ---------------------------------------------------------------------------
extern "C" void kernel_launch(void* const* d_in, const int* in_sizes, int n_in,
                              void* d_out, int out_size, void* d_ws, size_t ws_size,
                              hipStream_t stream) {
  (void)in_sizes; (void)n_in; (void)out_size; (void)ws_size;
  const int*   node_ids = (const int*)d_in[0];
  const int*   time_ids = (const int*)d_in[1];
  const float* adj      = (const float*)d_in[2];
  const int*   batch    = (const int*)d_in[3];
  float*       out      = (float*)d_out;

  char* ws = (char*)d_ws;
  unsigned long long* cells = (unsigned long long*)ws;        // 192 * 8B
  int*      meta  = (int*)(ws + 2048);                        // 64 * 3 * 4B
  uint32_t* nbits = (uint32_t*)(ws + 4096);                   // 64 * 64 * 4B

  // Host-side reproduction of kp1, kp2, kn = jax.random.split(key(42), 3):
  // threefry over counts [0..5] with key (0, 42); out = concat(r0s, r1s) over the 3 count pairs.
  uint32_t r00, r10, r01, r11, r02, r12;
  threefry2x32(0u, 42u, 0u, 3u, r00, r10);
  threefry2x32(0u, 42u, 1u, 4u, r01, r11);
  threefry2x32(0u, 42u, 2u, 5u, r02, r12);
  const uint32_t kp1a = r00, kp1b = r01;   // (out[0], out[1])
  const uint32_t kp2a = r02, kp2b = r10;   // (out[2], out[3])
  const uint32_t kna  = r11, knb  = r12;   // (out[4], out[5])

  sampler_prep<<<BATCH, 64, 0, stream>>>(node_ids, time_ids, adj, batch,
                                         meta, nbits, cells);
  dim3 grid(N_TOTAL / CHUNK, BATCH);
  sampler_main<<<grid, THREADS, 0, stream>>>(node_ids, time_ids, meta, nbits, cells,
                                             kp1a, kp1b, kp2a, kp2b, kna, knb);
  sampler_fin<<<1, BATCH * 3, 0, stream>>>(cells, out);
}